// Gener_embedding_50002009260273
// MI455X (gfx1250) — compile-verified
//
#include <hip/hip_runtime.h>
#include <stdint.h>

// ---- problem constants (from reference) ----
#define ROUTEID_NUM  100000
#define PAD_ID       (ROUTEID_NUM + 1)   // 100001
#define EMBED_SIZE   128
#define EMBED_VEC4   (EMBED_SIZE / 4)    // 32 float4 per row -> one wave32 per token
#define TOK_PER_BLK  256                 // tokens staged per block via TDM
#define WAVES_PER_BLK 8                  // 256 threads / wave32

// native vector types (nontemporal/TDM builtins need real vectors)
typedef float        f32x4 __attribute__((ext_vector_type(4)));
typedef unsigned int u32x4 __attribute__((ext_vector_type(4)));
typedef int          i32x8 __attribute__((ext_vector_type(8)));
typedef int          i32x4 __attribute__((ext_vector_type(4)));

// ---------------------------------------------------------------------------
// Per block: TDM (tensor_load_to_lds) stages this block's 256 token ids into
// LDS (one async 1KB DMA, issued by wave 0, tracked by TENSORcnt). After the
// barrier, each of the 8 waves moves one 512B embedding row per pass:
//   id from LDS (ds_load broadcast) -> road_map gather (scale_offset) ->
//   emb row global_load_b128 (L2-resident table) -> global_store_b128 th:NT.
// HBM traffic: 256MB NT writes + ~4MB reads => ~11us floor at 23.3 TB/s.
// ---------------------------------------------------------------------------
__global__ __launch_bounds__(256) void emb_gather_tdm_kernel(
    const int*   __restrict__ data,      // [n_tokens] road ids
    const int*   __restrict__ road_map,  // [ROUTEID_NUM+2] road -> cluster
    const f32x4* __restrict__ emb4,      // [CLUSTER_NUM * EMBED_VEC4]
    f32x4*       __restrict__ out4,      // [n_tokens * EMBED_VEC4]
    int n_tokens)
{
    __shared__ int s_ids[TOK_PER_BLK];

    const int base = blockIdx.x * TOK_PER_BLK;
    const int T    = min(TOK_PER_BLK, n_tokens - base);   // uniform

    // ---- wave 0: issue the TDM id-stage and wait for it -------------------
    if (threadIdx.x < 32) {
        const uint64_t gaddr   = (uint64_t)(uintptr_t)(data + base); // byte addr
        const uint32_t lds_off = (uint32_t)(uintptr_t)s_ids;         // LDS byte offset

        // D# group 0: count=1 (valid), gather_mode=0, lds_addr, global_addr[56:0], type=2
        u32x4 g0 = { 0x1u,
                     lds_off,
                     (uint32_t)gaddr,
                     (uint32_t)((gaddr >> 32) & 0x01FFFFFFu) | 0x80000000u };

        // D# group 1: workgroup_mask=0, data_size=2 (4B), no barrier/iterate/pad,
        //             tensor_dim0=T (bits 79:48), tensor_dim1=1 (bits 111:80),
        //             tile_dim0=T (bits 127:112), tile_dim1=1, tile_dim2=0,
        //             tensor_dim0_stride=T (bits 207:160)
        i32x8 g1 = { (int)(2u << 16),          // data_size = 4 bytes
                     (int)((uint32_t)T << 16), // tensor_dim0[15:0]
                     (int)(1u << 16),          // tensor_dim0[31:16]=0 | tensor_dim1=1
                     (int)((uint32_t)T << 16), // tensor_dim1 hi=0 | tile_dim0=T
                     1,                        // tile_dim1=1, tile_dim2=0
                     (int)(uint32_t)T,         // tensor_dim0_stride low32
                     0, 0 };

        i32x4 gz4 = { 0, 0, 0, 0 };            // groups 2/3: dims 2+ unused
        i32x8 gz8 = { 0, 0, 0, 0, 0, 0, 0, 0 };

        __builtin_amdgcn_tensor_load_to_lds(g0, g1, gz4, gz4, gz8, /*cpol=*/0);
        __builtin_amdgcn_s_wait_tensorcnt(0);  // ids resident in LDS
    }
    __syncthreads();                           // publish LDS to all 8 waves

    // ---- gather + stream out: one wave32 per token per pass ---------------
    const int wave  = threadIdx.x >> 5;        // 0..7
    const int chunk = threadIdx.x & 31;        // lane's float4 within the row

#pragma unroll 4
    for (int p = 0; p < TOK_PER_BLK / WAVES_PER_BLK; ++p) {
        const int tl    = p * WAVES_PER_BLK + wave;  // wave-uniform
        const int token = base + tl;
        if (token >= n_tokens) break;

        const int id = s_ids[tl];              // ds_load_b32, broadcast in-wave
        f32x4 v = (f32x4){0.f, 0.f, 0.f, 0.f};
        if (id != PAD_ID) {                    // wave-uniform branch
            const int c = road_map[id];        // global_load_b32 scale_offset
            v = emb4[c * EMBED_VEC4 + chunk];  // global_load_b128 (L2-hot table)
        }
        __builtin_nontemporal_store(v, &out4[token * EMBED_VEC4 + chunk]); // b128 NT
    }
}

// ---------------------------------------------------------------------------
extern "C" void kernel_launch(void* const* d_in, const int* in_sizes, int n_in,
                              void* d_out, int out_size, void* d_ws, size_t ws_size,
                              hipStream_t stream)
{
    // setup_inputs() order: data [B*S*1] int, road_map [100002] int, emb_table [4096*128] f32
    const int*   data     = (const int*)  d_in[0];
    const int*   road_map = (const int*)  d_in[1];
    const f32x4* emb4     = (const f32x4*)d_in[2];
    f32x4*       out4     = (f32x4*)      d_out;

    const int n_tokens = in_sizes[0];                      // 128*4096 = 524288
    const int n_blocks = (n_tokens + TOK_PER_BLK - 1) / TOK_PER_BLK; // 2048

    hipLaunchKernelGGL(emb_gather_tdm_kernel, dim3(n_blocks), dim3(256), 0, stream,
                       data, road_map, emb4, out4, n_tokens);
}